// Net_51874615001781
// MI455X (gfx1250) — compile-verified
//
#include <hip/hip_runtime.h>
#include <math.h>

#define N_NODES   100000
#define N_EDGES   1600000
#define F_IN      128
#define HIDDEN    128
#define N_CLASSES 40

typedef float v2f __attribute__((ext_vector_type(2)));
typedef float v8f __attribute__((ext_vector_type(8)));

// Native hardware fp32 atomic add (global_atomic_add_f32, no CAS loop).
__device__ __forceinline__ void atomAddF32(float* p, float v) {
    unsafeAtomicAdd(p, v);
}

// ---------------------------------------------------------------- degree prep
__global__ void deg_init_kernel(float* __restrict__ deg) {
    int i = blockIdx.x * blockDim.x + threadIdx.x;
    if (i < N_NODES) deg[i] = 1.0f;   // self-loop contributes 1
}

__global__ void deg_accum_kernel(const int* __restrict__ ei, float* __restrict__ deg) {
    int e = blockIdx.x * blockDim.x + threadIdx.x;
    if (e < N_EDGES) atomAddF32(&deg[ei[N_EDGES + e]], 1.0f);
}

__global__ void dinv_kernel(float* __restrict__ deg) {
    int i = blockIdx.x * blockDim.x + threadIdx.x;
    if (i < N_NODES) deg[i] = rsqrtf(fmaxf(deg[i], 1.0f));
}

// ---------------------------------------------------------------- zero fill
__global__ void zero4_kernel(float4* __restrict__ p, int n4) {
    int i = blockIdx.x * blockDim.x + threadIdx.x;
    if (i < n4) p[i] = make_float4(0.f, 0.f, 0.f, 0.f);
}

// ---------------------------------------------------------------- GEMM1: h1 = x @ W1   (M=100000, N=128, K=128)
// block = 64 threads (2 waves); wave w covers columns [w*64, w*64+64); grid.x = 6250 M-tiles.
// WMMA f32 16x16x4 layouts (wave32):
//   A (16x4): lane m=lane&15, khalf=lane>>4; a[v] = A[m][khalf*2+v]
//   B (4x16): col=lane&15;                   b[v] = B[khalf*2+v][col]
//   C (16x16): vgpr r: lanes0-15 -> (M=r, N=lane), lanes16-31 -> (M=r+8, N=lane-16)
__global__ __launch_bounds__(64) void gemm1_kernel(const float* __restrict__ x,
                                                   const float* __restrict__ W1,
                                                   float* __restrict__ h1) {
    const int wave = threadIdx.x >> 5;
    const int lane = threadIdx.x & 31;
    const int m    = lane & 15;
    const int kh   = (lane >> 4) << 1;      // 0 or 2
    const int tileM = blockIdx.x;           // 6250 tiles, 100000 = 6250*16 exactly
    const int n0    = wave * 64;

    const float* arow = x + (size_t)(tileM * 16 + m) * F_IN + kh;
    v8f c0 = {}, c1 = {}, c2 = {}, c3 = {};

    for (int kk = 0; kk < F_IN; kk += 4) {
        v2f a = *(const v2f*)(arow + kk);                 // 8B-aligned
        const float* bp = W1 + (size_t)(kk + kh) * HIDDEN + n0 + m;
        v2f b0 = { bp[0],  bp[HIDDEN + 0]  };
        v2f b1 = { bp[16], bp[HIDDEN + 16] };
        v2f b2 = { bp[32], bp[HIDDEN + 32] };
        v2f b3 = { bp[48], bp[HIDDEN + 48] };
        c0 = __builtin_amdgcn_wmma_f32_16x16x4_f32(false, a, false, b0, (short)0, c0, false, false);
        c1 = __builtin_amdgcn_wmma_f32_16x16x4_f32(false, a, false, b1, (short)0, c1, false, false);
        c2 = __builtin_amdgcn_wmma_f32_16x16x4_f32(false, a, false, b2, (short)0, c2, false, false);
        c3 = __builtin_amdgcn_wmma_f32_16x16x4_f32(false, a, false, b3, (short)0, c3, false, false);
    }

    const int rbase = tileM * 16 + ((lane >> 4) << 3);    // +8 for upper half-wave
    float* o = h1 + (size_t)rbase * HIDDEN + n0 + m;
#pragma unroll
    for (int r = 0; r < 8; r++) {
        o[(size_t)r * HIDDEN +  0] = c0[r];
        o[(size_t)r * HIDDEN + 16] = c1[r];
        o[(size_t)r * HIDDEN + 32] = c2[r];
        o[(size_t)r * HIDDEN + 48] = c3[r];
    }
}

// ---------------------------------------------------------------- edge scatter, layer 1 (128 feats)
// one wave per edge; lane handles float4 chunk
__global__ __launch_bounds__(256) void scatter1_kernel(const int* __restrict__ ei,
                                                       const float* __restrict__ dinv,
                                                       const float* __restrict__ h1,
                                                       float* __restrict__ agg1) {
    int e    = (blockIdx.x * blockDim.x + threadIdx.x) >> 5;
    int lane = threadIdx.x & 31;
    if (e >= N_EDGES) return;
    int s = ei[e];
    int d = ei[N_EDGES + e];
    float nm = dinv[s] * dinv[d];
    float4 v = ((const float4*)(h1 + (size_t)s * HIDDEN))[lane];
    float* o = agg1 + (size_t)d * HIDDEN + lane * 4;
    atomAddF32(o + 0, v.x * nm);
    atomAddF32(o + 1, v.y * nm);
    atomAddF32(o + 2, v.z * nm);
    atomAddF32(o + 3, v.w * nm);
}

// ---------------------------------------------------------------- self-loop + bias + relu (in place on agg1)
__global__ void relu_bias_kernel(float* __restrict__ agg1,
                                 const float* __restrict__ h1,
                                 const float* __restrict__ dinv,
                                 const float* __restrict__ b1) {
    int idx = blockIdx.x * blockDim.x + threadIdx.x;
    if (idx >= N_NODES * HIDDEN) return;
    int i = idx >> 7;
    int f = idx & 127;
    float di = dinv[i];
    float v  = agg1[idx] + h1[idx] * di * di + b1[f];
    agg1[idx] = fmaxf(v, 0.0f);
}

// ---------------------------------------------------------------- GEMM2: h2 = a1 @ W2  (N=40, padded to 48 tiles)
// block = 64 (2 waves); wave handles its own M tile; grid.x = 3125 (2 tiles/block)
__global__ __launch_bounds__(64) void gemm2_kernel(const float* __restrict__ a1,
                                                   const float* __restrict__ W2,
                                                   float* __restrict__ h2) {
    const int wave = threadIdx.x >> 5;
    const int lane = threadIdx.x & 31;
    const int m    = lane & 15;
    const int kh   = (lane >> 4) << 1;
    const int tileM = blockIdx.x * 2 + wave;              // 6250 tiles exactly

    const float* arow = a1 + (size_t)(tileM * 16 + m) * HIDDEN + kh;
    v8f c0 = {}, c1 = {}, c2 = {};
    const bool tail = (m < 8);                            // cols 32..39 valid, 40..47 padded

    for (int kk = 0; kk < HIDDEN; kk += 4) {
        v2f a = *(const v2f*)(arow + kk);
        const float* bp = W2 + (size_t)(kk + kh) * N_CLASSES + m;
        v2f b0 = { bp[0],  bp[N_CLASSES + 0]  };
        v2f b1 = { bp[16], bp[N_CLASSES + 16] };
        v2f b2 = tail ? (v2f){ bp[32], bp[N_CLASSES + 32] } : (v2f){ 0.f, 0.f };
        c0 = __builtin_amdgcn_wmma_f32_16x16x4_f32(false, a, false, b0, (short)0, c0, false, false);
        c1 = __builtin_amdgcn_wmma_f32_16x16x4_f32(false, a, false, b1, (short)0, c1, false, false);
        c2 = __builtin_amdgcn_wmma_f32_16x16x4_f32(false, a, false, b2, (short)0, c2, false, false);
    }

    const int rbase = tileM * 16 + ((lane >> 4) << 3);
    float* o = h2 + (size_t)rbase * N_CLASSES + m;
#pragma unroll
    for (int r = 0; r < 8; r++) {
        o[(size_t)r * N_CLASSES +  0] = c0[r];
        o[(size_t)r * N_CLASSES + 16] = c1[r];
        if (tail) o[(size_t)r * N_CLASSES + 32] = c2[r];
    }
}

// ---------------------------------------------------------------- edge scatter, layer 2 (40 feats)
__global__ __launch_bounds__(256) void scatter2_kernel(const int* __restrict__ ei,
                                                       const float* __restrict__ dinv,
                                                       const float* __restrict__ h2,
                                                       float* __restrict__ agg2) {
    int e    = (blockIdx.x * blockDim.x + threadIdx.x) >> 5;
    int lane = threadIdx.x & 31;
    if (e >= N_EDGES) return;
    int s = ei[e];
    int d = ei[N_EDGES + e];
    float nm = dinv[s] * dinv[d];
    const float* hs = h2 + (size_t)s * N_CLASSES;
    float* o = agg2 + (size_t)d * N_CLASSES;
    atomAddF32(o + lane, hs[lane] * nm);
    if (lane < 8) atomAddF32(o + 32 + lane, hs[32 + lane] * nm);
}

// ---------------------------------------------------------------- self-loop + bias + log_softmax
__global__ __launch_bounds__(256) void final_kernel(const float* __restrict__ agg2,
                                                    const float* __restrict__ h2,
                                                    const float* __restrict__ dinv,
                                                    const float* __restrict__ b2,
                                                    float* __restrict__ out) {
    int node = (blockIdx.x * blockDim.x + threadIdx.x) >> 5;
    int lane = threadIdx.x & 31;
    if (node >= N_NODES) return;
    float di = dinv[node];
    float d2 = di * di;
    const float* ag = agg2 + (size_t)node * N_CLASSES;
    const float* hh = h2   + (size_t)node * N_CLASSES;

    float v0 = ag[lane] + hh[lane] * d2 + b2[lane];
    float v1 = -INFINITY;
    if (lane < 8) v1 = ag[32 + lane] + hh[32 + lane] * d2 + b2[32 + lane];

    float mx = fmaxf(v0, v1);
#pragma unroll
    for (int off = 16; off; off >>= 1) mx = fmaxf(mx, __shfl_xor(mx, off, 32));
    float sv = expf(v0 - mx) + ((lane < 8) ? expf(v1 - mx) : 0.0f);
#pragma unroll
    for (int off = 16; off; off >>= 1) sv += __shfl_xor(sv, off, 32);
    float lse = mx + logf(sv);

    float* o = out + (size_t)node * N_CLASSES;
    o[lane] = v0 - lse;
    if (lane < 8) o[32 + lane] = v1 - lse;
}

// ---------------------------------------------------------------- launcher
extern "C" void kernel_launch(void* const* d_in, const int* in_sizes, int n_in,
                              void* d_out, int out_size, void* d_ws, size_t ws_size,
                              hipStream_t stream) {
    const float* x  = (const float*)d_in[0];
    const int*   ei = (const int*)  d_in[1];
    const float* W1 = (const float*)d_in[2];
    const float* b1 = (const float*)d_in[3];
    const float* W2 = (const float*)d_in[4];
    const float* b2 = (const float*)d_in[5];
    float* out = (float*)d_out;

    // workspace layout (aliased; ~103 MB):
    char* ws = (char*)d_ws;
    float* deg  = (float*)ws;                               // N floats -> becomes dinv in place
    float* h1   = (float*)(ws + 400128);                    // N*128 floats (51.2 MB)
    float* agg1 = h1 + (size_t)N_NODES * HIDDEN;            // N*128 floats (51.2 MB)
    float* h2   = h1;                                       // alias: a1 lives in agg1 while gemm2 writes h1 region
    float* agg2 = agg1;                                     // alias: a1 no longer needed after gemm2

    // 1) degree / norm
    deg_init_kernel <<<(N_NODES + 255) / 256, 256, 0, stream>>>(deg);
    deg_accum_kernel<<<(N_EDGES + 255) / 256, 256, 0, stream>>>(ei, deg);
    dinv_kernel     <<<(N_NODES + 255) / 256, 256, 0, stream>>>(deg);

    // 2) h1 = x @ W1
    gemm1_kernel<<<N_NODES / 16, 64, 0, stream>>>(x, W1, h1);

    // 3) agg1 = scatter(norm * h1[src] -> dst)
    {
        int n4 = (N_NODES * HIDDEN) / 4;
        zero4_kernel<<<(n4 + 255) / 256, 256, 0, stream>>>((float4*)agg1, n4);
    }
    scatter1_kernel<<<(N_EDGES * 32) / 256, 256, 0, stream>>>(ei, deg, h1, agg1);

    // 4) a1 = relu(agg1 + self-loop + b1)  (in place)
    relu_bias_kernel<<<(N_NODES * HIDDEN + 255) / 256, 256, 0, stream>>>(agg1, h1, deg, b1);

    // 5) h2 = a1 @ W2
    gemm2_kernel<<<N_NODES / 32, 64, 0, stream>>>(agg1, W2, h2);

    // 6) agg2 = scatter(norm * h2[src] -> dst)
    {
        int n4 = (N_NODES * N_CLASSES) / 4;
        zero4_kernel<<<(n4 + 255) / 256, 256, 0, stream>>>((float4*)agg2, n4);
    }
    scatter2_kernel<<<(N_EDGES * 32) / 256, 256, 0, stream>>>(ei, deg, h2, agg2);

    // 7) out = log_softmax(agg2 + self-loop + b2)
    final_kernel<<<(N_NODES * 32 + 255) / 256, 256, 0, stream>>>(agg2, h2, deg, b2, out);
}